// GINLayer_42820823941374
// MI455X (gfx1250) — compile-verified
//
#include <hip/hip_runtime.h>
#include <hip/hip_bf16.h>

#define N_NODES 100000
#define D 256
#define H 512            // 2*D hidden
#define BM 32            // rows per workgroup in MLP kernel
#define APAD (D + 4)     // padded LDS row strides (16B-aligned rows, conflict-free columns)
#define HPAD (H + 4)
constexpr float BN_EPS = 1e-5f;

typedef __attribute__((ext_vector_type(2))) float v2f;
typedef __attribute__((ext_vector_type(8))) float v8f;

// ---------------------------------------------------------------------------
// K0: h0 = (1+eps) * node_feats   (also zeroes the first 256 floats of the BN stat buffer)
// ---------------------------------------------------------------------------
__global__ void k_init(const float4* __restrict__ x, const float* __restrict__ eps,
                       float4* __restrict__ h0, float* __restrict__ stats) {
  const float s = 1.0f + eps[0];
  long i = (long)blockIdx.x * blockDim.x + threadIdx.x;
  const long total = (long)N_NODES * D / 4;
  if (i < total) {
    float4 v = x[i];
    v.x *= s; v.y *= s; v.z *= s; v.w *= s;
    h0[i] = v;
  }
  if (blockIdx.x == 0 && threadIdx.x < 2 * D) stats[threadIdx.x] = 0.0f;
}

// ---------------------------------------------------------------------------
// K1: edge scatter-add.  One wave32 per edge; indices forced scalar (SMEM),
// each lane moves 8 floats (2x float4 gather, 8x global_atomic_add_f32).
// node_feats + h0 total 2x102 MB.
// ---------------------------------------------------------------------------
__global__ void k_scatter(const float4* __restrict__ x, const int* __restrict__ src,
                          const int* __restrict__ dst, float* __restrict__ h0, int E) {
  long t = (long)blockIdx.x * blockDim.x + threadIdx.x;
  int edge = __builtin_amdgcn_readfirstlane((int)(t >> 5));  // wave-uniform
  int lane = threadIdx.x & 31;
  if (edge >= E) return;
  int s = __builtin_amdgcn_readfirstlane(src[edge]);
  int d = __builtin_amdgcn_readfirstlane(dst[edge]);
  const float4* xs = x + (long)s * (D / 4);
  float* hd = h0 + (long)d * D;
#pragma unroll
  for (int p = 0; p < 2; ++p) {
    int c4 = lane + p * 32;          // float4 index within the row, 0..63
    float4 v = xs[c4];
    atomicAdd(hd + c4 * 4 + 0, v.x);
    atomicAdd(hd + c4 * 4 + 1, v.y);
    atomicAdd(hd + c4 * 4 + 2, v.z);
    atomicAdd(hd + c4 * 4 + 3, v.w);
  }
}

// ---------------------------------------------------------------------------
// K2: fused MLP  out = relu(h0 @ W1 + b1) @ W2 + b2  via V_WMMA_F32_16X16X4_F32.
// BM=32 row strip per block; hidden strip (32x512) lives entirely in (dynamic)
// LDS.  8 waves; each wave drives TWO independent accumulators per K-loop
// (tiles ct and ct+8) sharing one A-fragment -> breaks the WMMA RAW chain.
// ---------------------------------------------------------------------------
__global__ __launch_bounds__(256) void k_mlp(const float* __restrict__ h0,
                                             const float* __restrict__ W1,
                                             const float* __restrict__ b1,
                                             const float* __restrict__ W2,
                                             const float* __restrict__ b2,
                                             float* __restrict__ out) {
  extern __shared__ float smem[];
  float* Asm = smem;                  // BM x APAD
  float* Hsm = smem + BM * APAD;      // BM x HPAD
#define ASM(r, c) Asm[(r) * APAD + (c)]
#define HSM(r, c) Hsm[(r) * HPAD + (c)]

  const int tid = threadIdx.x;
  const int row0 = blockIdx.x * BM;

  // Stage the 32x256 input strip into LDS with b128s (clamped tail rows).
  for (int i = tid; i < BM * (D / 4); i += 256) {
    int r = i / (D / 4), c4 = i % (D / 4);
    int gr = row0 + r;
    if (gr >= N_NODES) gr = N_NODES - 1;
    *(float4*)&ASM(r, c4 * 4) = ((const float4*)h0)[(long)gr * (D / 4) + c4];
  }
  __syncthreads();

  const int wave = tid >> 5;          // 0..7
  const int lane = tid & 31;
  const int lm = lane & 15;           // row (A) / col (B,C/D) within tile
  const int lk = (lane >> 4) * 2;     // K pair: lanes 0-15 -> K=0,1 ; 16-31 -> K=2,3
  const int mhi = (lane >> 4) * 8;    // C/D: VGPR i holds rows i and i+8

  // -------- Phase 1: Hsm = relu(A @ W1 + b1) --------
  // 2 row-tiles x 32 col-tiles; wave w does col pairs (w,w+8),(w+16,w+24).
  for (int rt = 0; rt < BM / 16; ++rt) {
    const float* Arow = &ASM(rt * 16 + lm, 0);
    for (int ct0 = wave; ct0 < H / 16; ct0 += 16) {
      const int colA = ct0 * 16, colB = (ct0 + 8) * 16;
      const float bvA = b1[colA + lm], bvB = b1[colB + lm];
      v8f c0, c1;
#pragma unroll
      for (int i = 0; i < 8; ++i) { c0[i] = bvA; c1[i] = bvB; }
      for (int k = 0; k < D; k += 4) {
        v2f a, bA, bB;
        a[0] = Arow[k + lk];
        a[1] = Arow[k + lk + 1];
        const float* Wk0 = W1 + (long)(k + lk) * H;
        const float* Wk1 = W1 + (long)(k + lk + 1) * H;
        bA[0] = Wk0[colA + lm]; bA[1] = Wk1[colA + lm];
        bB[0] = Wk0[colB + lm]; bB[1] = Wk1[colB + lm];
        c0 = __builtin_amdgcn_wmma_f32_16x16x4_f32(false, a, false, bA,
                                                   (short)0, c0, false, false);
        c1 = __builtin_amdgcn_wmma_f32_16x16x4_f32(false, a, false, bB,
                                                   (short)0, c1, false, false);
      }
#pragma unroll
      for (int i = 0; i < 8; ++i) {
        int m = rt * 16 + mhi + i;
        HSM(m, colA + lm) = c0[i] > 0.0f ? c0[i] : 0.0f;   // ReLU
        HSM(m, colB + lm) = c1[i] > 0.0f ? c1[i] : 0.0f;
      }
    }
  }
  __syncthreads();

  // -------- Phase 2: out = Hsm @ W2 + b2 --------
  // 2 row-tiles x 16 col-tiles; wave w does col pair (w, w+8).
  for (int rt = 0; rt < BM / 16; ++rt) {
    const float* Arow = &HSM(rt * 16 + lm, 0);
    const int ct0 = wave;
    const int colA = ct0 * 16, colB = (ct0 + 8) * 16;
    const float bvA = b2[colA + lm], bvB = b2[colB + lm];
    v8f c0, c1;
#pragma unroll
    for (int i = 0; i < 8; ++i) { c0[i] = bvA; c1[i] = bvB; }
    for (int k = 0; k < H; k += 4) {
      v2f a, bA, bB;
      a[0] = Arow[k + lk];
      a[1] = Arow[k + lk + 1];
      const float* Wk0 = W2 + (long)(k + lk) * D;
      const float* Wk1 = W2 + (long)(k + lk + 1) * D;
      bA[0] = Wk0[colA + lm]; bA[1] = Wk1[colA + lm];
      bB[0] = Wk0[colB + lm]; bB[1] = Wk1[colB + lm];
      c0 = __builtin_amdgcn_wmma_f32_16x16x4_f32(false, a, false, bA,
                                                 (short)0, c0, false, false);
      c1 = __builtin_amdgcn_wmma_f32_16x16x4_f32(false, a, false, bB,
                                                 (short)0, c1, false, false);
    }
#pragma unroll
    for (int i = 0; i < 8; ++i) {
      int gr = row0 + rt * 16 + mhi + i;
      if (gr < N_NODES) {
        out[(long)gr * D + colA + lm] = c0[i];
        out[(long)gr * D + colB + lm] = c1[i];
      }
    }
  }
#undef ASM
#undef HSM
}

// ---------------------------------------------------------------------------
// K3: BN batch stats — per-channel sum / sumsq partials, f32 atomics into ws.
// Thread j handles channel j (coalesced column reads), 512-row chunks/block.
// ---------------------------------------------------------------------------
__global__ void k_bnstats(const float* __restrict__ h2, float* __restrict__ stats, int n) {
  const int j = threadIdx.x;
  int r0 = blockIdx.x * 512;
  int r1 = r0 + 512; if (r1 > n) r1 = n;
  float s = 0.0f, s2 = 0.0f;
  for (int r = r0; r < r1; ++r) {
    float v = h2[(long)r * D + j];
    s += v;
    s2 += v * v;
  }
  atomicAdd(&stats[j], s);
  atomicAdd(&stats[D + j], s2);
}

// ---------------------------------------------------------------------------
// K4: BN apply (training mode, biased variance): out = out*scale + shift
// ---------------------------------------------------------------------------
__global__ void k_bnapply(float* __restrict__ h2, const float* __restrict__ stats,
                          const float* __restrict__ gamma, const float* __restrict__ beta,
                          int n) {
  const int j = threadIdx.x;
  const float inv_n = 1.0f / (float)n;
  const float mean = stats[j] * inv_n;
  const float var = stats[D + j] * inv_n - mean * mean;
  const float scale = gamma[j] * rsqrtf(var + BN_EPS);
  const float shift = beta[j] - mean * scale;
  for (int r = blockIdx.x; r < n; r += gridDim.x) {
    long idx = (long)r * D + j;
    h2[idx] = h2[idx] * scale + shift;
  }
}

// ---------------------------------------------------------------------------
extern "C" void kernel_launch(void* const* d_in, const int* in_sizes, int n_in,
                              void* d_out, int out_size, void* d_ws, size_t ws_size,
                              hipStream_t stream) {
  const float* x     = (const float*)d_in[0];
  const int*   src   = (const int*)d_in[1];
  const int*   dst   = (const int*)d_in[2];
  const float* eps   = (const float*)d_in[3];
  const float* W1    = (const float*)d_in[4];
  const float* b1    = (const float*)d_in[5];
  const float* W2    = (const float*)d_in[6];
  const float* b2    = (const float*)d_in[7];
  const float* gamma = (const float*)d_in[8];
  const float* beta  = (const float*)d_in[9];
  float* out = (float*)d_out;
  const int E = in_sizes[1];

  float* h0    = (float*)d_ws;                       // N x D accumulator (102.4 MB)
  float* stats = h0 + (long)N_NODES * D;             // sum[256] ++ sumsq[256]

  // K0: init accumulator with (1+eps)*x, zero stats
  k_init<<<(N_NODES * D / 4 + 255) / 256, 256, 0, stream>>>(
      (const float4*)x, eps, (float4*)h0, stats);

  // K1: scatter-add all edges (one wave32 per edge)
  k_scatter<<<(E + 7) / 8, 256, 0, stream>>>((const float4*)x, src, dst, h0, E);

  // K2: fused MLP with f32 WMMA, hidden layer kept in (dynamic) LDS
  const size_t smem_bytes = (size_t)(BM * APAD + BM * HPAD) * sizeof(float); // ~99 KB
  k_mlp<<<(N_NODES + BM - 1) / BM, 256, smem_bytes, stream>>>(h0, W1, b1, W2, b2, out);

  // K3/K4: BatchNorm training-mode stats + apply
  k_bnstats<<<(N_NODES + 511) / 512, 256, 0, stream>>>(out, stats, N_NODES);
  k_bnapply<<<1024, 256, 0, stream>>>(out, stats, gamma, beta, N_NODES);
}